// Critic_53626961658075
// MI455X (gfx1250) — compile-verified
//
#include <hip/hip_runtime.h>
#include <math.h>

typedef __attribute__((ext_vector_type(2))) float v2f;
typedef __attribute__((ext_vector_type(8))) float v8f;

#define VP 256          // padded feature dim (200 -> 256)
#define BATCH 1024
#define NTOK 256

// ---------------- ws layout (float offsets) ----------------
#define OFF_M      0                    // 256   : Mu(98) Md(98) Mp(49)
#define OFF_UDP    256                  // 3*1024*200
#define OFF_STATS  614656               // 600*2 (mean, rstd)
#define OFF_X      615856               // 4 * 1024*256
#define OFF_H      1664432              // 4 * 1024*256
#define OFF_W      2713008              // 4 * 256*256  (stored TRANSPOSED: Wt[n][k])
#define OFF_B      2975152              // 4 * 256

// ---------------------------------------------------------------------------
// Kernel 0: M = Wq @ Wk^T for the three heads (7x14, 7x14, 7x7)
// ---------------------------------------------------------------------------
__global__ void critic_mmat(const float* __restrict__ uq, const float* __restrict__ uk,
                            const float* __restrict__ dq, const float* __restrict__ dk,
                            const float* __restrict__ pq, const float* __restrict__ pk,
                            float* __restrict__ Mmat) {
    int t = threadIdx.x;
    if (t < 98) {
        int i = t / 14, c = t % 14; float s = 0.f;
        for (int v = 0; v < 200; ++v) s += uq[i*200+v] * uk[c*200+v];
        Mmat[t] = s;
    } else if (t < 196) {
        int u = t - 98; int i = u / 14, c = u % 14; float s = 0.f;
        for (int v = 0; v < 200; ++v) s += dq[i*200+v] * dk[c*200+v];
        Mmat[t] = s;
    } else if (t < 245) {
        int u = t - 196; int i = u / 7, c = u % 7; float s = 0.f;
        for (int v = 0; v < 200; ++v) s += pq[i*200+v] * pk[c*200+v];
        Mmat[t] = s;
    }
}

// ---------------------------------------------------------------------------
// Kernel 1: zero-pad the four 200x200 weights to 256x256, stored TRANSPOSED
//           (Wt[layer][n][k]) so WMMA B-fragments are contiguous float2 loads
// ---------------------------------------------------------------------------
__global__ void critic_wprep(const float* __restrict__ tW1, const float* __restrict__ tb1,
                             const float* __restrict__ eW2, const float* __restrict__ eb2,
                             float* __restrict__ Wt, float* __restrict__ bp) {
    int blk = blockIdx.x;          // 4*256 blocks: (layer, k-row)
    int layer = blk >> 8, row = blk & 255;
    int col = threadIdx.x;
    float w = 0.f;
    if (row < 200 && col < 200)
        w = (layer < 3) ? tW1[((size_t)layer*200 + row)*200 + col] : eW2[(size_t)row*200 + col];
    Wt[((size_t)layer*VP + col)*VP + row] = w;     // transposed store
    if (row == 0) {
        float b = 0.f;
        if (col < 200) b = (layer < 3) ? tb1[layer*200 + col] : eb2[col];
        bp[layer*VP + col] = b;
    }
}

// ---------------------------------------------------------------------------
// Kernel 2: per-batch attention (folded) + first e-MLP layer
// ---------------------------------------------------------------------------
__global__ void __launch_bounds__(256) critic_attn(
    const float* __restrict__ merged, const float* __restrict__ a,
    const float* __restrict__ Mmat,
    const float* __restrict__ uWv, const float* __restrict__ dWv, const float* __restrict__ pWv,
    const float* __restrict__ eW1, const float* __restrict__ eb1,
    float* __restrict__ U, float* __restrict__ D, float* __restrict__ P,
    float* __restrict__ X3) {

    const int b = blockIdx.x;
    const int tid = threadIdx.x;

    __shared__ float feat[NTOK][16];
    __shared__ float su[NTOK], sd[NTOK], sp[NTOK];
    __shared__ float red[NTOK];
    __shared__ float wsum[48];

    // coalesced stage of the 256x15 token tile
    const float* src = merged + (size_t)b * NTOK * 15;
    for (int i = tid; i < NTOK * 15; i += 256) {
        int n = i / 15, c = i - n * 15;
        feat[n][c] = src[i];
    }
    __syncthreads();

    float subj_id  = feat[0][0];
    float subj_loc = feat[0][2];
    __syncthreads();                      // all read subj before mutation

    feat[tid][0] -= subj_id;
    feat[tid][7] -= subj_id;
    if (tid == 0) feat[0][6] = a[b];
    __syncthreads();

    float ego[7];
    #pragma unroll
    for (int i = 0; i < 7; ++i) ego[i] = feat[0][i];

    // ego @ M   (tiny, recomputed per thread)
    float eMu[14], eMd[14], eMp[7];
    #pragma unroll
    for (int c = 0; c < 14; ++c) {
        float s0 = 0.f, s1 = 0.f;
        #pragma unroll
        for (int i = 0; i < 7; ++i) { s0 += ego[i]*Mmat[i*14+c]; s1 += ego[i]*Mmat[98+i*14+c]; }
        eMu[c] = s0; eMd[c] = s1;
    }
    #pragma unroll
    for (int c = 0; c < 7; ++c) {
        float s2 = 0.f;
        #pragma unroll
        for (int i = 0; i < 7; ++i) s2 += ego[i]*Mmat[196+i*7+c];
        eMp[c] = s2;
    }

    const float scale = 0.070710678118654752f;   // 1/sqrt(200)
    {
        int n = tid;
        float loc = feat[n][2], flag = feat[n][14];
        float s0 = 0.f, s1 = 0.f, s2 = 0.f;
        #pragma unroll
        for (int c = 0; c < 14; ++c) { float f = feat[n][c]; s0 += eMu[c]*f; s1 += eMd[c]*f; }
        #pragma unroll
        for (int c = 0; c < 7; ++c) s2 += eMp[c]*feat[n][c];
        bool um = (loc < subj_loc) && (flag == 1.0f);
        bool dm = (loc > subj_loc) && (flag == 1.0f);
        bool pm = (flag == 0.0f);
        su[n] = um ? s0*scale : -1e9f;
        sd[n] = dm ? s1*scale : -1e9f;
        sp[n] = pm ? s2*scale : -1e9f;
    }
    __syncthreads();

    auto reduceMax = [&](float* s) -> float {
        red[tid] = s[tid]; __syncthreads();
        for (int st = 128; st > 0; st >>= 1) { if (tid < st) red[tid] = fmaxf(red[tid], red[tid+st]); __syncthreads(); }
        float r = red[0]; __syncthreads(); return r;
    };
    auto reduceSum = [&](float* s) -> float {
        red[tid] = s[tid]; __syncthreads();
        for (int st = 128; st > 0; st >>= 1) { if (tid < st) red[tid] += red[tid+st]; __syncthreads(); }
        float r = red[0]; __syncthreads(); return r;
    };

    float mu = reduceMax(su), md = reduceMax(sd), mp = reduceMax(sp);
    bool any_u = mu > -1e8f, any_d = md > -1e8f, any_p = mp > -1e8f;

    su[tid] = expf(su[tid] - mu);
    sd[tid] = expf(sd[tid] - md);
    sp[tid] = expf(sp[tid] - mp);
    float sumu = reduceSum(su), sumd = reduceSum(sd), sump = reduceSum(sp);

    // normalized weighted feature sums (threads: head = tid/16, c = tid%16)
    if (tid < 48) {
        int head = tid >> 4, c = tid & 15;
        int nf = (head == 2) ? 7 : 14;
        if (c < nf) {
            float* sarr = (head == 0) ? su : (head == 1) ? sd : sp;
            float acc = 0.f;
            for (int n = 0; n < NTOK; ++n) acc += sarr[n] * feat[n][c];
            float denom = (head == 0) ? sumu : (head == 1) ? sumd : sump;
            wsum[tid] = acc / denom;
        } else wsum[tid] = 0.f;
    }
    __syncthreads();

    // 14 -> 200 matvec per head, plus e-MLP layer 1 into padded X3
    int j = tid;
    if (j < 200) {
        float au = 0.f, ad = 0.f, ap = 0.f;
        #pragma unroll
        for (int c = 0; c < 14; ++c) { au += wsum[c]*uWv[c*200+j]; ad += wsum[16+c]*dWv[c*200+j]; }
        #pragma unroll
        for (int c = 0; c < 7; ++c)  ap += wsum[32+c]*pWv[c*200+j];
        U[(size_t)b*200 + j] = any_u ? au : 0.f;
        D[(size_t)b*200 + j] = any_d ? ad : 0.f;
        P[(size_t)b*200 + j] = any_p ? ap : 0.f;
    }
    float q1 = 0.f;
    if (j < 200) {
        float s = eb1[j];
        #pragma unroll
        for (int i = 0; i < 4; ++i) s += feat[0][3+i] * eW1[i*200+j];
        q1 = fmaxf(s, 0.f);
    }
    X3[(size_t)b*VP + j] = q1;
}

// ---------------------------------------------------------------------------
// Kernel 3: BN statistics over the batch dimension (600 columns)
// ---------------------------------------------------------------------------
__global__ void critic_stats(const float* __restrict__ UDP, float* __restrict__ stats) {
    int hb = blockIdx.x;                 // head*200 + col
    int head = hb / 200, col = hb - head*200;
    const float* X = UDP + (size_t)head * BATCH * 200;
    int tid = threadIdx.x;
    float s = 0.f, s2 = 0.f;
    for (int r = tid; r < BATCH; r += 256) { float v = X[(size_t)r*200 + col]; s += v; s2 += v*v; }
    __shared__ float rs[256], rq[256];
    rs[tid] = s; rq[tid] = s2; __syncthreads();
    for (int st = 128; st > 0; st >>= 1) { if (tid < st) { rs[tid] += rs[tid+st]; rq[tid] += rq[tid+st]; } __syncthreads(); }
    if (tid == 0) {
        float mean = rs[0] * (1.f/BATCH);
        float var  = rq[0] * (1.f/BATCH) - mean*mean;
        stats[hb*2]   = mean;
        stats[hb*2+1] = rsqrtf(var + 1e-5f);
    }
}

// ---------------------------------------------------------------------------
// Kernel 4: normalize + zero-pad into GEMM inputs X[0..2]
// ---------------------------------------------------------------------------
__global__ void critic_bn(const float* __restrict__ UDP, const float* __restrict__ stats,
                          const float* __restrict__ gamma, const float* __restrict__ beta,
                          float* __restrict__ X) {
    int blk = blockIdx.x;                // 3*1024 blocks: (head, b)
    int head = blk >> 10, b = blk & 1023;
    int j = threadIdx.x;
    float out = 0.f;
    if (j < 200) {
        float v = UDP[(size_t)head*BATCH*200 + (size_t)b*200 + j];
        float mean = stats[(head*200+j)*2], rstd = stats[(head*200+j)*2 + 1];
        out = gamma[j]*(v - mean)*rstd + beta[j];
    }
    X[(size_t)head*BATCH*VP + (size_t)b*VP + j] = out;
}

// ---------------------------------------------------------------------------
// Kernel 5: 4 layers of (1024x256)@(256x256) fp32 WMMA GEMM + bias + act
//   grid (64 row-tiles, 1, 4 layers); 4 waves/block; each wave computes a
//   16x64 stripe (4 fp32 accumulators) -> 1 A-frag reused by 4 WMMAs/k-step.
//   W is pre-transposed so both A and B fragments are contiguous float2.
// ---------------------------------------------------------------------------
__global__ void __launch_bounds__(128) critic_gemm(
    const float* __restrict__ Xall, const float* __restrict__ Wtall,
    const float* __restrict__ ball, float* __restrict__ Hall) {

    const int layer = blockIdx.z;
    const int row0  = blockIdx.x * 16;
    const int wave  = threadIdx.x >> 5;
    const int lane  = threadIdx.x & 31;
    const int col0  = wave * 64;            // this wave's 64-column stripe

    const float* X    = Xall  + (size_t)layer * BATCH * VP;
    const float* Wt   = Wtall + (size_t)layer * VP * VP;
    const float* bias = ball  + layer * VP;
    float*       H    = Hall  + (size_t)layer * BATCH * VP;

    const int hi  = lane >> 4;              // half-wave selects K pair
    const int l15 = lane & 15;

    // A frag (16x4 f32): lanes 0-15 -> K={k,k+1}, lanes 16-31 -> K={k+2,k+3}
    const float* aptr = X + (size_t)(row0 + l15) * VP + (hi ? 2 : 0);
    // B frags from transposed W: Wt[n][k] -> contiguous in k per lane
    const float* bp0 = Wt + (size_t)(col0 +  0 + l15) * VP + (hi ? 2 : 0);
    const float* bp1 = Wt + (size_t)(col0 + 16 + l15) * VP + (hi ? 2 : 0);
    const float* bp2 = Wt + (size_t)(col0 + 32 + l15) * VP + (hi ? 2 : 0);
    const float* bp3 = Wt + (size_t)(col0 + 48 + l15) * VP + (hi ? 2 : 0);

    v8f acc0 = {}, acc1 = {}, acc2 = {}, acc3 = {};
    #pragma unroll 4
    for (int k = 0; k < VP; k += 4) {
        v2f av, b0, b1, b2, b3;
        av.x = aptr[k]; av.y = aptr[k + 1];
        b0.x = bp0[k];  b0.y = bp0[k + 1];
        b1.x = bp1[k];  b1.y = bp1[k + 1];
        b2.x = bp2[k];  b2.y = bp2[k + 1];
        b3.x = bp3[k];  b3.y = bp3[k + 1];
        acc0 = __builtin_amdgcn_wmma_f32_16x16x4_f32(false, av, false, b0, (short)0, acc0, false, false);
        acc1 = __builtin_amdgcn_wmma_f32_16x16x4_f32(false, av, false, b1, (short)0, acc1, false, false);
        acc2 = __builtin_amdgcn_wmma_f32_16x16x4_f32(false, av, false, b2, (short)0, acc2, false, false);
        acc3 = __builtin_amdgcn_wmma_f32_16x16x4_f32(false, av, false, b3, (short)0, acc3, false, false);
    }

    // epilogue: C/D layout — VGPR r holds (M=r, M=r+8) split at half-wave
    v8f accs[4] = { acc0, acc1, acc2, acc3 };
    #pragma unroll
    for (int t = 0; t < 4; ++t) {
        int c0 = col0 + t * 16;
        float bcol = bias[c0 + l15];
        #pragma unroll
        for (int r = 0; r < 8; ++r) {
            int row = row0 + r + (hi ? 8 : 0);
            float v = accs[t][r] + bcol;
            v = (layer < 3) ? (v > 0.f ? v : (expf(v) - 1.f))   // elu
                            : fmaxf(v, 0.f);                    // relu
            H[(size_t)row * VP + c0 + l15] = v;
        }
    }
}

// ---------------------------------------------------------------------------
// Kernel 6: final 200->1 dots + bias sums; one wave per batch row
// ---------------------------------------------------------------------------
__global__ void critic_final(const float* __restrict__ Hall,
                             const float* __restrict__ tW2, const float* __restrict__ tb2,
                             const float* __restrict__ eW3, const float* __restrict__ eb3,
                             float* __restrict__ out) {
    int wave = threadIdx.x >> 5, lane = threadIdx.x & 31;
    int b = blockIdx.x * 8 + wave;
    float a0 = 0.f, a1 = 0.f, a2 = 0.f, a3 = 0.f;
    const size_t L = (size_t)BATCH * VP;
    for (int j = lane; j < 200; j += 32) {
        float h0 = Hall[0*L + (size_t)b*VP + j];
        float h1 = Hall[1*L + (size_t)b*VP + j];
        float h2 = Hall[2*L + (size_t)b*VP + j];
        float h3 = Hall[3*L + (size_t)b*VP + j];
        a0 += h0 * tW2[j];
        a1 += h1 * tW2[200 + j];
        a2 += h2 * tW2[400 + j];
        a3 += h3 * eW3[j];
    }
    float t = a0 + a1 + a2 + a3;
    for (int off = 16; off > 0; off >>= 1) t += __shfl_down(t, off, 32);
    if (lane == 0) out[b] = t + tb2[0] + tb2[1] + tb2[2] + eb3[0];
}

// ---------------------------------------------------------------------------
extern "C" void kernel_launch(void* const* d_in, const int* in_sizes, int n_in,
                              void* d_out, int out_size, void* d_ws, size_t ws_size,
                              hipStream_t stream) {
    const float* merged = (const float*)d_in[0];
    const float* a      = (const float*)d_in[1];
    const float* up_Wq  = (const float*)d_in[2];
    const float* up_Wk  = (const float*)d_in[3];
    const float* up_Wv  = (const float*)d_in[4];
    const float* dn_Wq  = (const float*)d_in[5];
    const float* dn_Wk  = (const float*)d_in[6];
    const float* dn_Wv  = (const float*)d_in[7];
    const float* pv_Wq  = (const float*)d_in[8];
    const float* pv_Wk  = (const float*)d_in[9];
    const float* pv_Wv  = (const float*)d_in[10];
    const float* t_W1   = (const float*)d_in[11];
    const float* t_b1   = (const float*)d_in[12];
    const float* t_W2   = (const float*)d_in[13];
    const float* t_b2   = (const float*)d_in[14];
    const float* e_W1   = (const float*)d_in[15];
    const float* e_b1   = (const float*)d_in[16];
    const float* e_W2   = (const float*)d_in[17];
    const float* e_b2   = (const float*)d_in[18];
    const float* e_W3   = (const float*)d_in[19];
    const float* e_b3   = (const float*)d_in[20];
    const float* gamma  = (const float*)d_in[21];
    const float* beta   = (const float*)d_in[22];

    float* ws    = (float*)d_ws;
    float* Mmat  = ws + OFF_M;
    float* UDP   = ws + OFF_UDP;
    float* stats = ws + OFF_STATS;
    float* X     = ws + OFF_X;
    float* H     = ws + OFF_H;
    float* Wt    = ws + OFF_W;
    float* bp    = ws + OFF_B;

    critic_mmat<<<1, 256, 0, stream>>>(up_Wq, up_Wk, dn_Wq, dn_Wk, pv_Wq, pv_Wk, Mmat);
    critic_wprep<<<4 * 256, 256, 0, stream>>>(t_W1, t_b1, e_W2, e_b2, Wt, bp);
    critic_attn<<<BATCH, 256, 0, stream>>>(merged, a, Mmat, up_Wv, dn_Wv, pv_Wv,
                                           e_W1, e_b1,
                                           UDP, UDP + (size_t)BATCH*200, UDP + (size_t)2*BATCH*200,
                                           X + (size_t)3*BATCH*VP);
    critic_stats<<<600, 256, 0, stream>>>(UDP, stats);
    critic_bn<<<3 * BATCH, 256, 0, stream>>>(UDP, stats, gamma, beta, X);
    critic_gemm<<<dim3(64, 1, 4), 128, 0, stream>>>(X, Wt, bp, H);
    critic_final<<<BATCH / 8, 256, 0, stream>>>(H, t_W2, t_b2, e_W3, e_b3, (float*)d_out);
}